// LSTM_75539884802382
// MI455X (gfx1250) — compile-verified
//
#include <hip/hip_runtime.h>
#include <math.h>

// ---------------------------------------------------------------------------
// LSTM inference (3 layers, H=2048, IN=512, batch=1) for gfx1250 / MI455X.
// HBM-bandwidth-bound mat-vec (~352 MB fp32 weights/call, 23.3 TB/s -> ~15us
// floor). fp32 WMMA 16x16x4 used as a full-precision dot engine; weights
// streamed with b128 loads (512 B per wave per load instruction).
// ---------------------------------------------------------------------------

typedef __attribute__((ext_vector_type(2))) float v2f;
typedef __attribute__((ext_vector_type(4))) float v4f;
typedef __attribute__((ext_vector_type(8))) float v8f;

#define HSZ   2048
#define GATES (4 * HSZ)   // 8192 gate rows per layer

// One wave per gate row r: g[r] = dot(W_ih[r], x) + dot(W_hh[r], h) + b_ih + b_hh.
// Diagonal accumulation over 128-element chunks:
//   lane l<16  holds W[r][c0+8l .. +3], lane l+16 holds W[r][c0+8l+4 .. +7]
//   WMMA0 uses elements {0,1,4,5} of each 8-group, WMMA1 uses {2,3,6,7};
//   A and B (x) use the identical mapping, so D[m][m] accumulates disjoint
//   exact fp32 partial dot products.
__global__ __launch_bounds__(256) void lstm_gates_kernel(
    const float* __restrict__ W_ih, const float* __restrict__ W_hh,
    const float* __restrict__ b_ih, const float* __restrict__ b_hh,
    const float* __restrict__ x,    const float* __restrict__ h,
    float* __restrict__ g, int in_dim) {

  const int lane = threadIdx.x & 31;
  const int wave = threadIdx.x >> 5;
  const int row  = blockIdx.x * (blockDim.x >> 5) + wave;   // 0..8191

  // float4 offset inside a 128-element chunk (contiguous 512B per wave).
  const int koff = 8 * (lane & 15) + 4 * (lane >> 4);

  v8f acc = {};  // C/D accumulator (16x16 f32); we use only the diagonal

  const float* wr = W_ih + (size_t)row * (size_t)in_dim;
  for (int c0 = 0; c0 < in_dim; c0 += 128) {
    v4f a = *(const v4f*)(wr + c0 + koff);   // global_load_b128, coalesced
    v4f b = *(const v4f*)(x  + c0 + koff);   // same pattern on x (cache-hot)
    v2f a0 = {a.x, a.y}, a1 = {a.z, a.w};
    v2f b0 = {b.x, b.y}, b1 = {b.z, b.w};
    acc = __builtin_amdgcn_wmma_f32_16x16x4_f32(false, a0, false, b0,
                                                (short)0, acc, false, false);
    acc = __builtin_amdgcn_wmma_f32_16x16x4_f32(false, a1, false, b1,
                                                (short)0, acc, false, false);
  }

  const float* wh = W_hh + (size_t)row * (size_t)HSZ;
  for (int c0 = 0; c0 < HSZ; c0 += 128) {
    v4f a = *(const v4f*)(wh + c0 + koff);
    v4f b = *(const v4f*)(h  + c0 + koff);
    v2f a0 = {a.x, a.y}, a1 = {a.z, a.w};
    v2f b0 = {b.x, b.y}, b1 = {b.z, b.w};
    acc = __builtin_amdgcn_wmma_f32_16x16x4_f32(false, a0, false, b0,
                                                (short)0, acc, false, false);
    acc = __builtin_amdgcn_wmma_f32_16x16x4_f32(false, a1, false, b1,
                                                (short)0, acc, false, false);
  }

  // Extract diagonal D[m][m]: m<8 -> lane m, vgpr m ; m>=8 -> lane m+16, vgpr m-8.
  float d = 0.0f;
#pragma unroll
  for (int i = 0; i < 8; ++i) {
    if (lane == i)      d = acc[i];
    if (lane == i + 24) d = acc[i];
  }
  // Wave32 tree reduction.
#pragma unroll
  for (int off = 16; off > 0; off >>= 1) d += __shfl_xor(d, off, 32);

  if (lane == 0) g[row] = d + b_ih[row] + b_hh[row];
}

__device__ __forceinline__ float sigm(float v) { return 1.0f / (1.0f + expf(-v)); }

// Elementwise cell update: PyTorch gate order i, f, g, o in blocks of H.
__global__ __launch_bounds__(256) void lstm_cell_kernel(
    const float* __restrict__ g, const float* __restrict__ c_old,
    float* __restrict__ h_ws, float* __restrict__ h_out, float* __restrict__ c_out) {
  int j = blockIdx.x * blockDim.x + threadIdx.x;   // 0..2047
  float gi = g[j];
  float gf = g[HSZ + j];
  float gg = g[2 * HSZ + j];
  float go = g[3 * HSZ + j];
  float c_new = sigm(gf) * c_old[j] + sigm(gi) * tanhf(gg);
  float h_new = sigm(go) * tanhf(c_new);
  h_ws[j]  = h_new;   // aligned scratch copy (feeds next layer's gates)
  h_out[j] = h_new;   // d_out new_h section
  c_out[j] = c_new;   // d_out new_s section
}

// y = dot(W_out, h2) + b_out  (single-block reduction)
__global__ __launch_bounds__(256) void out_proj_kernel(
    const float* __restrict__ W, const float* __restrict__ b,
    const float* __restrict__ h, float* __restrict__ y) {
  __shared__ float red[256];
  float s = 0.0f;
  for (int j = threadIdx.x; j < HSZ; j += 256) s += W[j] * h[j];
  red[threadIdx.x] = s;
  __syncthreads();
#pragma unroll
  for (int k = 128; k > 0; k >>= 1) {
    if (threadIdx.x < (unsigned)k) red[threadIdx.x] += red[threadIdx.x + k];
    __syncthreads();
  }
  if (threadIdx.x == 0) y[0] = red[0] + b[0];
}

extern "C" void kernel_launch(void* const* d_in, const int* in_sizes, int n_in,
                              void* d_out, int out_size, void* d_ws, size_t ws_size,
                              hipStream_t stream) {
  (void)in_sizes; (void)n_in; (void)out_size; (void)ws_size;

  const float* x     = (const float*)d_in[0];   // [512]
  const float* h_old = (const float*)d_in[1];   // [3,2048]
  const float* s_old = (const float*)d_in[2];   // [3,2048]
  const float* Wih[3] = {(const float*)d_in[3],  (const float*)d_in[7],  (const float*)d_in[11]};
  const float* Whh[3] = {(const float*)d_in[4],  (const float*)d_in[8],  (const float*)d_in[12]};
  const float* bih[3] = {(const float*)d_in[5],  (const float*)d_in[9],  (const float*)d_in[13]};
  const float* bhh[3] = {(const float*)d_in[6],  (const float*)d_in[10], (const float*)d_in[14]};
  const float* W_out = (const float*)d_in[15];  // [2048]
  const float* b_out = (const float*)d_in[16];  // [1]

  float* out   = (float*)d_out;
  float* y     = out;                 // [1]
  float* new_h = out + 1;             // [3*2048]
  float* new_s = out + 1 + 3 * HSZ;   // [3*2048]

  float* g    = (float*)d_ws;         // [8192] gate pre-activations
  float* hcur = g + GATES;            // [2048] current layer input (16B aligned)

  const float* cur_in = x;
  int in_dim = 512;
  for (int l = 0; l < 3; ++l) {
    // 8192 rows, 1 wave per row, 8 waves (256 threads) per block -> 1024 blocks.
    lstm_gates_kernel<<<dim3(GATES / 8), dim3(256), 0, stream>>>(
        Wih[l], Whh[l], bih[l], bhh[l], cur_in, h_old + l * HSZ, g, in_dim);
    lstm_cell_kernel<<<dim3(HSZ / 256), dim3(256), 0, stream>>>(
        g, s_old + l * HSZ, hcur, new_h + l * HSZ, new_s + l * HSZ);
    cur_in = hcur;
    in_dim = HSZ;
  }
  out_proj_kernel<<<dim3(1), dim3(256), 0, stream>>>(W_out, b_out, hcur, y);
}